// GcnNet_18107582120631
// MI455X (gfx1250) — compile-verified
//
#include <hip/hip_runtime.h>

typedef __attribute__((ext_vector_type(2))) float v2f;
typedef __attribute__((ext_vector_type(8))) float v8f;

#define HIDDEN 128

__device__ __forceinline__ void atomAddF32(float* p, float v) {
    __hip_atomic_fetch_add(p, v, __ATOMIC_RELAXED, __HIP_MEMORY_SCOPE_AGENT);
}

__device__ __forceinline__ v8f wmma_f32(v2f a, v2f b, v8f c) {
    return __builtin_amdgcn_wmma_f32_16x16x4_f32(
        false, a, false, b, (short)0, c, false, false);
}

// ---------------------------------------------------------------------------
// Zero a float buffer (grid-stride).
// ---------------------------------------------------------------------------
__global__ void zero_kernel(float* __restrict__ p, long n) {
    long i = (long)blockIdx.x * blockDim.x + threadIdx.x;
    long stride = (long)gridDim.x * blockDim.x;
    for (; i < n; i += stride) p[i] = 0.0f;
}

// ---------------------------------------------------------------------------
// EmbeddingBag(mode='sum') with per-sample weights, + bias + ReLU.
// One block (128 threads) per node; thread d owns hidden dim d.
// ---------------------------------------------------------------------------
__global__ void embed_bag_kernel(const int* __restrict__ ids,
                                 const int* __restrict__ offsets,
                                 const float* __restrict__ psw,
                                 const float* __restrict__ emb,
                                 const float* __restrict__ bias,
                                 float* __restrict__ h) {
    const int node = blockIdx.x;
    const int d = threadIdx.x;                 // 0..127
    const int beg = offsets[node];
    const int end = offsets[node + 1];
    float acc = 0.0f;
    for (int j = beg; j < end; ++j) {
        const int id = ids[j];
        const float w = psw[j];
        acc = fmaf(emb[(long)id * HIDDEN + d], w, acc);
    }
    acc += bias[d];
    h[(long)node * HIDDEN + d] = fmaxf(acc, 0.0f);
}

// ---------------------------------------------------------------------------
// Edge gather + dual scatter-add: one wave32 per edge, lane owns 4 dims.
//   ppi_acc[dst] += h[src] * ppi_w[e];  res_acc[dst] += h[src] * self_w[e]
// ---------------------------------------------------------------------------
__global__ void edge_scatter_kernel(const int* __restrict__ src,
                                    const int* __restrict__ dst,
                                    const float* __restrict__ ppi_w,
                                    const float* __restrict__ self_w,
                                    const float* __restrict__ h,
                                    float* __restrict__ ppi_acc,
                                    float* __restrict__ res_acc,
                                    int n_edges) {
    const int e = (int)(((long)blockIdx.x * blockDim.x + threadIdx.x) >> 5);
    const int lane = threadIdx.x & 31;
    if (e >= n_edges) return;
    const int s = src[e];
    const int t = dst[e];
    const float wp = ppi_w[e];
    const float ws = self_w[e];
    const float4 hv = *(const float4*)(h + (long)s * HIDDEN + lane * 4);
    float* pp = ppi_acc + (long)t * HIDDEN + lane * 4;
    float* rp = res_acc + (long)t * HIDDEN + lane * 4;
    atomAddF32(pp + 0, hv.x * wp);
    atomAddF32(pp + 1, hv.y * wp);
    atomAddF32(pp + 2, hv.z * wp);
    atomAddF32(pp + 3, hv.w * wp);
    atomAddF32(rp + 0, hv.x * ws);
    atomAddF32(rp + 1, hv.y * ws);
    atomAddF32(rp + 2, hv.z * ws);
    atomAddF32(rp + 3, hv.w * ws);
}

// ---------------------------------------------------------------------------
// Hidden GEMM, register-blocked 1 M-tile x 4 N-tiles per wave:
//   H = relu(A @ W + bias) + res      A:[N,128]  W:[128,128]
// One A b64 load feeds 4 V_WMMA_F32_16X16X4_F32 per K-step.
// ---------------------------------------------------------------------------
__global__ void gemm_hidden_kernel(const float* __restrict__ A,
                                   const float* __restrict__ W,
                                   const float* __restrict__ bias,
                                   const float* __restrict__ res,
                                   float* __restrict__ H,
                                   int n_nodes) {
    const int lane = threadIdx.x & 31;
    const int wave = threadIdx.x >> 5;
    const int mtiles  = (n_nodes + 15) >> 4;
    const int ngroups = HIDDEN >> 6;                     // 2 groups of 64 cols
    const int tile = blockIdx.x * (blockDim.x >> 5) + wave;
    if (tile >= mtiles * ngroups) return;
    const int mt = tile / ngroups;
    const int ng = tile - mt * ngroups;
    const int m0 = mt << 4;
    const int n0 = ng << 6;
    const int col  = lane & 15;
    const int half = lane >> 4;

    int ar = m0 + col; if (ar >= n_nodes) ar = n_nodes - 1;
    const float* arow = A + (size_t)ar * HIDDEN;
    const float* wbase = W + n0 + col;

    v8f acc0 = {}, acc1 = {}, acc2 = {}, acc3 = {};
#pragma unroll 4
    for (int k = 0; k < HIDDEN; k += 4) {
        v2f a = *(const v2f*)(arow + k + 2 * half);
        const float* wr0 = wbase + (size_t)(k + 2 * half) * HIDDEN;
        const float* wr1 = wr0 + HIDDEN;
        v2f b0, b1, b2, b3;
        b0.x = wr0[0];  b0.y = wr1[0];
        b1.x = wr0[16]; b1.y = wr1[16];
        b2.x = wr0[32]; b2.y = wr1[32];
        b3.x = wr0[48]; b3.y = wr1[48];
        acc0 = wmma_f32(a, b0, acc0);
        acc1 = wmma_f32(a, b1, acc1);
        acc2 = wmma_f32(a, b2, acc2);
        acc3 = wmma_f32(a, b3, acc3);
    }

    const float bv0 = bias[n0 + col];
    const float bv1 = bias[n0 + 16 + col];
    const float bv2 = bias[n0 + 32 + col];
    const float bv3 = bias[n0 + 48 + col];

    if (m0 + 16 <= n_nodes) {            // uniform fast path (always for 50000)
#pragma unroll
        for (int i = 0; i < 8; ++i) {
            const size_t idx = (size_t)(m0 + i + 8 * half) * HIDDEN + n0 + col;
            H[idx +  0] = fmaxf(acc0[i] + bv0, 0.0f) + res[idx +  0];
            H[idx + 16] = fmaxf(acc1[i] + bv1, 0.0f) + res[idx + 16];
            H[idx + 32] = fmaxf(acc2[i] + bv2, 0.0f) + res[idx + 32];
            H[idx + 48] = fmaxf(acc3[i] + bv3, 0.0f) + res[idx + 48];
        }
    } else {
#pragma unroll
        for (int i = 0; i < 8; ++i) {
            const int m = m0 + i + 8 * half;
            if (m < n_nodes) {
                const size_t idx = (size_t)m * HIDDEN + n0 + col;
                H[idx +  0] = fmaxf(acc0[i] + bv0, 0.0f) + res[idx +  0];
                H[idx + 16] = fmaxf(acc1[i] + bv1, 0.0f) + res[idx + 16];
                H[idx + 32] = fmaxf(acc2[i] + bv2, 0.0f) + res[idx + 32];
                H[idx + 48] = fmaxf(acc3[i] + bv3, 0.0f) + res[idx + 48];
            }
        }
    }
}

// ---------------------------------------------------------------------------
// Output GEMM, register-blocked 1 M-tile x 4 N-tiles per wave:
//   out = H @ W_out + b_out          H:[N,128]  W_out:[128,n_classes]
// Ragged N edge handled by CLAMPING B column indices (keeps EXEC uniform
// through the WMMA loop); only stores are guarded.
// ---------------------------------------------------------------------------
__global__ void gemm_out_kernel(const float* __restrict__ A,
                                const float* __restrict__ W,
                                const float* __restrict__ bias,
                                float* __restrict__ out,
                                int n_nodes, int n_classes) {
    const int lane = threadIdx.x & 31;
    const int wave = threadIdx.x >> 5;
    const int ntiles  = (n_classes + 15) >> 4;
    const int ngroups = (ntiles + 3) >> 2;
    const int mtiles  = (n_nodes + 15) >> 4;
    const int tile = blockIdx.x * (blockDim.x >> 5) + wave;
    if (tile >= mtiles * ngroups) return;
    const int mt = tile / ngroups;
    const int ng = tile - mt * ngroups;
    const int m0 = mt << 4;
    const int n0 = ng << 6;
    const int col  = lane & 15;
    const int half = lane >> 4;

    int ar = m0 + col; if (ar >= n_nodes) ar = n_nodes - 1;
    const float* arow = A + (size_t)ar * HIDDEN;

    // Clamped column indices (safe loads, garbage only in never-stored cols)
    int c0 = n0 +  0 + col; if (c0 >= n_classes) c0 = n_classes - 1;
    int c1 = n0 + 16 + col; if (c1 >= n_classes) c1 = n_classes - 1;
    int c2 = n0 + 32 + col; if (c2 >= n_classes) c2 = n_classes - 1;
    int c3 = n0 + 48 + col; if (c3 >= n_classes) c3 = n_classes - 1;
    const float* w0 = W + c0;
    const float* w1 = W + c1;
    const float* w2 = W + c2;
    const float* w3 = W + c3;

    v8f acc0 = {}, acc1 = {}, acc2 = {}, acc3 = {};
#pragma unroll 4
    for (int k = 0; k < HIDDEN; k += 4) {
        v2f a = *(const v2f*)(arow + k + 2 * half);
        const size_t r0 = (size_t)(k + 2 * half) * n_classes;
        const size_t r1 = r0 + n_classes;
        v2f b0, b1, b2, b3;
        b0.x = w0[r0]; b0.y = w0[r1];
        b1.x = w1[r0]; b1.y = w1[r1];
        b2.x = w2[r0]; b2.y = w2[r1];
        b3.x = w3[r0]; b3.y = w3[r1];
        acc0 = wmma_f32(a, b0, acc0);
        acc1 = wmma_f32(a, b1, acc1);
        acc2 = wmma_f32(a, b2, acc2);
        acc3 = wmma_f32(a, b3, acc3);
    }

    const float bv0 = bias[c0];
    const float bv1 = bias[c1];
    const float bv2 = bias[c2];
    const float bv3 = bias[c3];
    const bool ok0 = (n0 +  0 + col) < n_classes;
    const bool ok1 = (n0 + 16 + col) < n_classes;
    const bool ok2 = (n0 + 32 + col) < n_classes;
    const bool ok3 = (n0 + 48 + col) < n_classes;

    if (m0 + 16 <= n_nodes && n0 + 64 <= n_classes) {   // uniform fast path
#pragma unroll
        for (int i = 0; i < 8; ++i) {
            float* o = out + (size_t)(m0 + i + 8 * half) * n_classes + n0 + col;
            o[ 0] = acc0[i] + bv0;
            o[16] = acc1[i] + bv1;
            o[32] = acc2[i] + bv2;
            o[48] = acc3[i] + bv3;
        }
    } else {
#pragma unroll
        for (int i = 0; i < 8; ++i) {
            const int m = m0 + i + 8 * half;
            if (m < n_nodes) {
                float* o = out + (size_t)m * n_classes + n0 + col;
                if (ok0) o[ 0] = acc0[i] + bv0;
                if (ok1) o[16] = acc1[i] + bv1;
                if (ok2) o[32] = acc2[i] + bv2;
                if (ok3) o[48] = acc3[i] + bv3;
            }
        }
    }
}

// ---------------------------------------------------------------------------
// Host launcher
// ---------------------------------------------------------------------------
extern "C" void kernel_launch(void* const* d_in, const int* in_sizes, int n_in,
                              void* d_out, int out_size, void* d_ws, size_t ws_size,
                              hipStream_t stream) {
    const int*   input_ids = (const int*)  d_in[0];
    const int*   offsets   = (const int*)  d_in[1];
    const int*   src       = (const int*)  d_in[2];
    const int*   dst       = (const int*)  d_in[3];
    const float* psw       = (const float*)d_in[4];
    const float* ppi_w     = (const float*)d_in[5];
    const float* self_w    = (const float*)d_in[6];
    const float* emb       = (const float*)d_in[7];
    const float* in_bias   = (const float*)d_in[8];
    const float* W_layers  = (const float*)d_in[9];
    const float* b_layers  = (const float*)d_in[10];
    const float* W_out     = (const float*)d_in[11];
    const float* b_out     = (const float*)d_in[12];

    const int n_nodes   = in_sizes[1] - 1;       // 50000
    const int n_edges   = in_sizes[2];           // 800000
    const int n_classes = in_sizes[12];          // 1000
    const int n_layers  = in_sizes[10] / HIDDEN; // 2

    float* h   = (float*)d_ws;
    float* ppi = h   + (size_t)n_nodes * HIDDEN;
    float* res = ppi + (size_t)n_nodes * HIDDEN;

    // h = relu(embedding_bag + bias)
    embed_bag_kernel<<<n_nodes, HIDDEN, 0, stream>>>(
        input_ids, offsets, psw, emb, in_bias, h);

    const long accum_elems = 2L * n_nodes * HIDDEN;   // ppi + res contiguous
    const int edge_blocks = (int)(((long)n_edges * 32 + 255) / 256);
    const int mtiles = (n_nodes + 15) >> 4;

    for (int l = 0; l < n_layers; ++l) {
        zero_kernel<<<1024, 256, 0, stream>>>(ppi, accum_elems);
        edge_scatter_kernel<<<edge_blocks, 256, 0, stream>>>(
            src, dst, ppi_w, self_w, h, ppi, res, n_edges);
        const int tiles = mtiles * (HIDDEN >> 6);     // 4 N-tiles per wave
        gemm_hidden_kernel<<<(tiles + 7) / 8, 256, 0, stream>>>(
            ppi, W_layers + (size_t)l * HIDDEN * HIDDEN,
            b_layers + (size_t)l * HIDDEN, res, h, n_nodes);
    }

    const int ntiles  = (n_classes + 15) >> 4;
    const int ngroups = (ntiles + 3) >> 2;
    const int tiles   = mtiles * ngroups;
    gemm_out_kernel<<<(tiles + 7) / 8, 256, 0, stream>>>(
        h, W_out, b_out, (float*)d_out, n_nodes, n_classes);
}